// ProcrustesLoss_15006615733302
// MI455X (gfx1250) — compile-verified
//
#include <hip/hip_runtime.h>
#include <stdint.h>

// ---------------- Problem constants (from reference) ----------------
#define NJ   17            // joints
#define CPJ  (NJ * 3)      // 51 floats per batch item
#define TPB  128           // threads per block = batch items per block (4 waves)
#define CHUNK (TPB * CPJ)  // 6528 floats per tensor per block (26112 B)

typedef unsigned int       u32;
typedef unsigned long long u64;

typedef unsigned int v4u __attribute__((ext_vector_type(4)));
typedef int          v8i __attribute__((ext_vector_type(8)));
typedef int          v4i __attribute__((ext_vector_type(4)));
typedef float        v2f __attribute__((ext_vector_type(2)));
typedef float        v8f __attribute__((ext_vector_type(8)));

#if __has_builtin(__builtin_amdgcn_tensor_load_to_lds) && __has_builtin(__builtin_amdgcn_s_wait_tensorcnt)
#define HAS_TDM 1
#else
#define HAS_TDM 0
#endif

#if __has_builtin(__builtin_amdgcn_wmma_f32_16x16x4_f32)
#define HAS_WMMA_F32 1
#else
#define HAS_WMMA_F32 0
#endif

// Low 32 bits of a generic pointer into LDS == byte offset within the
// workgroup LDS allocation (addrspacecast AS3->generic keeps offset in LSBs).
__device__ __forceinline__ u32 lds_byte_offset(const void* p) {
  return (u32)(uintptr_t)p;
}

#if HAS_TDM
// 1-D contiguous global -> LDS DMA via the Tensor Data Mover.
// D# layout per CDNA5 ISA.
__device__ __forceinline__ void tdm_load_1d(const float* gsrc, u32 lds_off_bytes, u32 nelem) {
  u64 ga = (u64)(uintptr_t)gsrc;
  v4u g0;
  g0.x = 1u;                                            // count=1, user mode, no gather
  g0.y = lds_off_bytes;                                 // lds_addr [63:32]
  g0.z = (u32)(ga & 0xFFFFFFFFu);                       // global_addr [95:64]
  g0.w = (u32)((ga >> 32) & 0x01FFFFFFu) | (2u << 30);  // global_addr[56:32] | type=2

  v8i g1;
  g1[0] = (int)(2u << 16);                              // wg_mask=0 | data_size=2 (4B)
  g1[1] = (int)((nelem & 0xFFFFu) << 16);               // tensor_dim0 [15:0]
  g1[2] = (int)((nelem >> 16) | (1u << 16));            // tensor_dim0[31:16] | tensor_dim1=1
  g1[3] = (int)((nelem & 0xFFFFu) << 16);               // tensor_dim1 hi=0 | tile_dim0
  g1[4] = 1;                                            // tile_dim1=1 | tile_dim2=0
  g1[5] = (int)nelem;                                   // tensor_dim0_stride [31:0]
  g1[6] = 0;                                            // stride hi | dim1_stride lo
  g1[7] = 0;

  v4i gz = {0, 0, 0, 0};
#if defined(__clang_major__) && (__clang_major__ >= 23)
  v8i gz8 = {0, 0, 0, 0, 0, 0, 0, 0};
  __builtin_amdgcn_tensor_load_to_lds(g0, g1, gz, gz, gz8, 0);
#else
  __builtin_amdgcn_tensor_load_to_lds(g0, g1, gz, gz, 0);
#endif
}
#endif

// Jacobi rotation on symmetric k[3][3], accumulating rotations into v[3][3].
// p,q are literal constants so all indexing stays in registers.
#define JROT(p, q)                                                              \
  {                                                                             \
    float apq = k[p][q];                                                        \
    if (__builtin_fabsf(apq) > 1e-30f) {                                        \
      float tau = (k[q][q] - k[p][p]) / (2.0f * apq);                           \
      float t_ = (tau >= 0.0f ? 1.0f : -1.0f) /                                 \
                 (__builtin_fabsf(tau) + __builtin_sqrtf(1.0f + tau * tau));    \
      float c_ = 1.0f / __builtin_sqrtf(1.0f + t_ * t_);                        \
      float s_ = t_ * c_;                                                       \
      _Pragma("unroll") for (int r = 0; r < 3; r++) {                           \
        float a0 = k[r][p], a1 = k[r][q];                                       \
        k[r][p] = c_ * a0 - s_ * a1;                                            \
        k[r][q] = s_ * a0 + c_ * a1;                                            \
      }                                                                         \
      _Pragma("unroll") for (int r = 0; r < 3; r++) {                           \
        float a0 = k[p][r], a1 = k[q][r];                                       \
        k[p][r] = c_ * a0 - s_ * a1;                                            \
        k[q][r] = s_ * a0 + c_ * a1;                                            \
      }                                                                         \
      _Pragma("unroll") for (int r = 0; r < 3; r++) {                           \
        float a0 = v[r][p], a1 = v[r][q];                                       \
        v[r][p] = c_ * a0 - s_ * a1;                                            \
        v[r][q] = s_ * a0 + c_ * a1;                                            \
      }                                                                         \
    }                                                                           \
  }

#define CSWAP(i, j)                                                             \
  if (w##i < w##j) {                                                            \
    float tw = w##i; w##i = w##j; w##j = tw;                                    \
    _Pragma("unroll") for (int r = 0; r < 3; r++) {                             \
      float tv = v[r][i]; v[r][i] = v[r][j]; v[r][j] = tv;                      \
    }                                                                           \
  }

__global__ __launch_bounds__(TPB) void procrustes_main(
    const float* __restrict__ pred, const float* __restrict__ targ,
    float* __restrict__ partial, int B) {
  __shared__ float sm[2 * CHUNK];
  __shared__ float red[TPB / 32];

  const int tid = threadIdx.x;
  const int blk = blockIdx.x;
  const long long base = (long long)blk * CHUNK;
  const long long total = (long long)B * CPJ;
  long long rem = total - base;
  const int elems = rem > (long long)CHUNK ? CHUNK : (int)rem;

#if HAS_TDM
  if (tid == 0) {
    u32 lds0 = lds_byte_offset(&sm[0]);
    tdm_load_1d(pred + base, lds0, (u32)elems);
    tdm_load_1d(targ + base, lds0 + (u32)(CHUNK * sizeof(float)), (u32)elems);
  }
  __builtin_amdgcn_s_wait_tensorcnt(0);
  __syncthreads();
#else
  for (int i = tid; i < elems; i += TPB) {
    sm[i] = pred[base + i];
    sm[CHUNK + i] = targ[base + i];
  }
  __syncthreads();
#endif

  const int b = blk * TPB + tid;
  float lane_sum = 0.0f;

  if (b < B) {
    const float* P = &sm[tid * CPJ];
    const float* T = &sm[CHUNK + tid * CPJ];

    // ---- pass 1: means ----
    float mpx = 0.f, mpy = 0.f, mpz = 0.f, mtx = 0.f, mty = 0.f, mtz = 0.f;
#pragma unroll
    for (int j = 0; j < NJ; j++) {
      mpx += P[3 * j + 0]; mpy += P[3 * j + 1]; mpz += P[3 * j + 2];
      mtx += T[3 * j + 0]; mty += T[3 * j + 1]; mtz += T[3 * j + 2];
    }
    const float invn = 1.0f / (float)NJ;
    mpx *= invn; mpy *= invn; mpz *= invn;
    mtx *= invn; mty *= invn; mtz *= invn;

    // ---- pass 2: norm sums + unscaled cross-covariance H ----
    float pn = 0.f, tn = 0.f;
    float h[3][3] = {{0.f, 0.f, 0.f}, {0.f, 0.f, 0.f}, {0.f, 0.f, 0.f}};
#pragma unroll
    for (int j = 0; j < NJ; j++) {
      float px = P[3 * j + 0] - mpx, py = P[3 * j + 1] - mpy, pz = P[3 * j + 2] - mpz;
      float tx = T[3 * j + 0] - mtx, ty = T[3 * j + 1] - mty, tz = T[3 * j + 2] - mtz;
      pn += __builtin_sqrtf(px * px + py * py + pz * pz);
      tn += __builtin_sqrtf(tx * tx + ty * ty + tz * tz);
      h[0][0] += px * tx; h[0][1] += px * ty; h[0][2] += px * tz;
      h[1][0] += py * tx; h[1][1] += py * ty; h[1][2] += py * tz;
      h[2][0] += pz * tx; h[2][1] += pz * ty; h[2][2] += pz * tz;
    }
    const float scale = tn / (pn + 1e-8f);
#pragma unroll
    for (int a = 0; a < 3; a++)
#pragma unroll
      for (int c = 0; c < 3; c++) h[a][c] *= scale;

    // ---- K = H^T H, Jacobi eigen-decomposition (right singular vectors) ----
    float k[3][3];
#pragma unroll
    for (int a = 0; a < 3; a++)
#pragma unroll
      for (int c = 0; c < 3; c++)
        k[a][c] = h[0][a] * h[0][c] + h[1][a] * h[1][c] + h[2][a] * h[2][c];
    float v[3][3] = {{1.f, 0.f, 0.f}, {0.f, 1.f, 0.f}, {0.f, 0.f, 1.f}};
#pragma unroll
    for (int sw = 0; sw < 4; sw++) { JROT(0, 1) JROT(0, 2) JROT(1, 2) }

    float w0 = k[0][0], w1 = k[1][1], w2 = k[2][2];
    CSWAP(0, 1) CSWAP(0, 2) CSWAP(1, 2)

    // ---- left singular vectors: u_i = H v_i / s_i, orthonormalized ----
    float s0 = __builtin_sqrtf(__builtin_fmaxf(w0, 0.f));
    float s1 = __builtin_sqrtf(__builtin_fmaxf(w1, 0.f));
    float i0 = 1.0f / __builtin_fmaxf(s0, 1e-12f);
    float i1 = 1.0f / __builtin_fmaxf(s1, 1e-12f);
    float u[3][3];
#pragma unroll
    for (int r = 0; r < 3; r++) {
      u[r][0] = (h[r][0] * v[0][0] + h[r][1] * v[1][0] + h[r][2] * v[2][0]) * i0;
      u[r][1] = (h[r][0] * v[0][1] + h[r][1] * v[1][1] + h[r][2] * v[2][1]) * i1;
    }
    {  // Gram–Schmidt safety on u0,u1; u2 = u0 x u1
      float n0 = u[0][0] * u[0][0] + u[1][0] * u[1][0] + u[2][0] * u[2][0];
      float r0 = 1.0f / __builtin_sqrtf(__builtin_fmaxf(n0, 1e-24f));
      u[0][0] *= r0; u[1][0] *= r0; u[2][0] *= r0;
      float d01 = u[0][0] * u[0][1] + u[1][0] * u[1][1] + u[2][0] * u[2][1];
      u[0][1] -= d01 * u[0][0]; u[1][1] -= d01 * u[1][0]; u[2][1] -= d01 * u[2][0];
      float n1 = u[0][1] * u[0][1] + u[1][1] * u[1][1] + u[2][1] * u[2][1];
      float r1 = 1.0f / __builtin_sqrtf(__builtin_fmaxf(n1, 1e-24f));
      u[0][1] *= r1; u[1][1] *= r1; u[2][1] *= r1;
      u[0][2] = u[1][0] * u[2][1] - u[2][0] * u[1][1];
      u[1][2] = u[2][0] * u[0][1] - u[0][0] * u[2][1];
      u[2][2] = u[0][0] * u[1][1] - u[1][0] * u[0][1];
    }

    // ---- R = Vh @ U^T : R[a][b] = sum_c V[c][a] * U[b][c] ----
    // Split off the c=2 term so the reference's reflection branch
    // (negate column 2 of Vh) is R_adj = A - Bm.
    float A_[3][3], Bm[3][3], R[3][3];
#pragma unroll
    for (int a = 0; a < 3; a++)
#pragma unroll
      for (int c = 0; c < 3; c++) {
        A_[a][c] = v[0][a] * u[c][0] + v[1][a] * u[c][1];
        Bm[a][c] = v[2][a] * u[c][2];
        R[a][c] = A_[a][c] + Bm[a][c];
      }
    float det = R[0][0] * (R[1][1] * R[2][2] - R[1][2] * R[2][1]) -
                R[0][1] * (R[1][0] * R[2][2] - R[1][2] * R[2][0]) +
                R[0][2] * (R[1][0] * R[2][1] - R[1][1] * R[2][0]);
    if (!(det > 0.0f)) {
#pragma unroll
      for (int a = 0; a < 3; a++)
#pragma unroll
        for (int c = 0; c < 3; c++) R[a][c] = A_[a][c] - Bm[a][c];
    }

    // ---- pass 3: aligned = pred_s @ R^T + mean_t ; accumulate per-joint L2 ----
#pragma unroll
    for (int j = 0; j < NJ; j++) {
      float px = (P[3 * j + 0] - mpx) * scale;
      float py = (P[3 * j + 1] - mpy) * scale;
      float pz = (P[3 * j + 2] - mpz) * scale;
      float ax = px * R[0][0] + py * R[0][1] + pz * R[0][2] + mtx - T[3 * j + 0];
      float ay = px * R[1][0] + py * R[1][1] + pz * R[1][2] + mty - T[3 * j + 1];
      float az = px * R[2][0] + py * R[2][1] + pz * R[2][2] + mtz - T[3 * j + 2];
      lane_sum += __builtin_sqrtf(ax * ax + ay * ay + az * az);
    }
  }

  // ---- block reduction (wave32 shuffles, then LDS) ----
#pragma unroll
  for (int m = 16; m > 0; m >>= 1) lane_sum += __shfl_xor(lane_sum, m, 32);
  const int wave = tid >> 5;
  if ((tid & 31) == 0) red[wave] = lane_sum;
  __syncthreads();
  if (tid == 0) {
    float s = 0.f;
#pragma unroll
    for (int i = 0; i < TPB / 32; i++) s += red[i];
    partial[blk] = s;
  }
}

// Final reduction: one wave32. Each lane folds its strided share of the block
// partials into the two A-operand registers of a V_WMMA_F32_16X16X4_F32, then
// D = A x ones(4x16) + 0 row-sums all 64 lane-partials in one matrix op
// (exact f32, layout-independent since B is all-ones):
//   total = sum_m D[m][0] = (8-reg sum on lane 0) + (8-reg sum on lane 16).
__global__ __launch_bounds__(32) void procrustes_reduce(
    const float* __restrict__ partial, int n, float* __restrict__ out, float inv_count) {
  const int lane = threadIdx.x;
  float a0 = 0.f, a1 = 0.f;
  for (int i = lane; i < n; i += 64) a0 += partial[i];
  for (int i = lane + 32; i < n; i += 64) a1 += partial[i];

#if HAS_WMMA_F32
  v2f A;
  A.x = a0;
  A.y = a1;
  v2f Bv;
  Bv.x = 1.0f;
  Bv.y = 1.0f;
  v8f C = {};
  v8f D = __builtin_amdgcn_wmma_f32_16x16x4_f32(
      /*neg_a=*/false, A, /*neg_b=*/false, Bv,
      /*c_mod=*/(short)0, C, /*reuse_a=*/false, /*reuse_b=*/false);
  float s = ((D[0] + D[1]) + (D[2] + D[3])) + ((D[4] + D[5]) + (D[6] + D[7]));
  float other = __shfl(s, 16, 32);  // lane16 holds rows 8..15 of column 0
  if (lane == 0) out[0] = (s + other) * inv_count;
#else
  float s = a0 + a1;
#pragma unroll
  for (int m = 16; m > 0; m >>= 1) s += __shfl_xor(s, m, 32);
  if (lane == 0) out[0] = s * inv_count;
#endif
}

extern "C" void kernel_launch(void* const* d_in, const int* in_sizes, int n_in,
                              void* d_out, int out_size, void* d_ws, size_t ws_size,
                              hipStream_t stream) {
  const float* pred = (const float*)d_in[0];
  const float* targ = (const float*)d_in[1];
  float* out = (float*)d_out;
  float* partial = (float*)d_ws;

  const int B = in_sizes[0] / CPJ;              // 262144 for the reference shapes
  const int nblocks = (B + TPB - 1) / TPB;      // 2048

  procrustes_main<<<nblocks, TPB, 0, stream>>>(pred, targ, partial, B);

  const float inv_count = 1.0f / ((float)B * (float)NJ);
  procrustes_reduce<<<1, 32, 0, stream>>>(partial, nblocks, out, inv_count);
}